// SingleInputPeptidePocketConvLayer_11072425689947
// MI455X (gfx1250) — compile-verified
//
#include <hip/hip_runtime.h>

// SingleInputPeptidePocketConvLayer for MI455X (gfx1250, wave32).
//
// Per sample b:
//   lp      = (int)x[b,0]
//   pep_pad = [zeros(1,20); x[b,1:301].reshape(15,20)]        (16 x 20)
//   Sel[p,r]= #{k : table[lp,p,k] == r}                        (34 x 16)
//   S       = Sel @ pep_pad                                    (34 x 20)  <-- WMMA f32 16x16x4
//   out[p,o]= sum_f S[p,o+f] * kernel[pocket[b,p], f]          (34 x 16)  <-- VALU tail
//
// Memory-bound problem (~176 MB total @ 23.3 TB/s); one wave per sample,
// coalesced row loads into LDS, coalesced 128B-per-instruction output stores.

typedef float v2f __attribute__((ext_vector_type(2)));
typedef float v8f __attribute__((ext_vector_type(8)));

#define LCONST 15
#define ACONST 20
#define FCONST 5
#define PCONST 34
#define OCONST 16
#define XSTRIDE 335          // 1 + 15*20 + 34
#define OUTSTRIDE 544        // 34 * 16
#define PEP_STRIDE 32        // padded LDS row stride (cols 20..31 zeroed)
#define WAVES_PER_BLOCK 8

__global__ __launch_bounds__(256) void pep_pocket_conv_kernel(
    const float* __restrict__ x,
    const float* __restrict__ kern,
    const int*   __restrict__ table,
    float*       __restrict__ out,
    int nB)
{
    __shared__ float ldsPep[WAVES_PER_BLOCK][16 * PEP_STRIDE]; // pep_pad, 16x32
    __shared__ float ldsS  [WAVES_PER_BLOCK][PCONST * ACONST]; // S, 34x20

    const int lane = threadIdx.x & 31;
    const int wave = threadIdx.x >> 5;
    const int half = lane >> 4;      // 0: lanes 0-15, 1: lanes 16-31
    const int lmod = lane & 15;

    const int  b      = blockIdx.x * WAVES_PER_BLOCK + wave;
    const bool active = (b < nB);
    const int  bsafe  = active ? b : 0;

    float* pep = ldsPep[wave];
    float* S   = ldsS[wave];

    const float* xr = x + (size_t)bsafe * XSTRIDE;

    // ---- Stage 1: pep_pad into LDS (row 0 + cols 20..31 stay zero) ----
    #pragma unroll
    for (int j = lane; j < 16 * PEP_STRIDE; j += 32) pep[j] = 0.0f;
    #pragma unroll
    for (int i = lane; i < LCONST * ACONST; i += 32) {
        float v = active ? xr[1 + i] : 0.0f;   // coalesced 128B per iteration
        int r = i / ACONST;                    // 0..14 -> pep_pad rows 1..15
        int c = i - r * ACONST;
        pep[(r + 1) * PEP_STRIDE + c] = v;
    }
    const int lp = active ? (int)xr[0] : 8;    // wave-uniform peptide length

    __syncthreads();   // publish pep_pad across lanes (uniform barrier)

    // ---- Stage 2: B fragments for S = Sel x pep_pad ----
    // Assumed f32 4x16 B layout (mirror of documented 16x4 A layout):
    //   VGPR j, lane-half h, lane n=lmod  ->  B[K = 4*kk + 2*h + j][N = nt*16 + n]
    v2f Bf[2][4];
    #pragma unroll
    for (int nt = 0; nt < 2; ++nt) {
        #pragma unroll
        for (int kk = 0; kk < 4; ++kk) {
            const int k0 = 4 * kk + 2 * half;
            const int n  = nt * 16 + lmod;
            v2f bb;
            bb[0] = pep[(k0 + 0) * PEP_STRIDE + n];
            bb[1] = pep[(k0 + 1) * PEP_STRIDE + n];
            Bf[nt][kk] = bb;
        }
    }

    // ---- Stage 3: WMMA — 3 M-tiles x 2 N-tiles x 4 K-steps ----
    #pragma unroll
    for (int t = 0; t < 3; ++t) {
        const int m = t * 16 + lmod;           // pocket row for this lane (A: M=lane)
        // Sel row as packed 4-bit counters: nibble r = count of table entries == r.
        // r=0 (padding) intentionally skipped -> Sel[:,0]=0 (pep_pad row 0 is zero anyway).
        unsigned long long cnt = 0ULL;
        if (m < PCONST) {
            const int* trow = table + ((size_t)lp * PCONST + m) * LCONST;
            #pragma unroll
            for (int s = 0; s < LCONST; ++s) {
                const int e = trow[s];                       // 0..15
                cnt += e ? (1ULL << (4 * e)) : 0ULL;
            }
        }
        #pragma unroll
        for (int nt = 0; nt < 2; ++nt) {
            v8f C = {};
            #pragma unroll
            for (int kk = 0; kk < 4; ++kk) {
                const int k0 = 4 * kk + 2 * half;            // A: VGPR j -> K = k0 + j
                v2f A;
                A[0] = (float)((cnt >> (4 * (k0 + 0))) & 0xFULL);
                A[1] = (float)((cnt >> (4 * (k0 + 1))) & 0xFULL);
                C = __builtin_amdgcn_wmma_f32_16x16x4_f32(
                        /*neg_a=*/false, A, /*neg_b=*/false, Bf[nt][kk],
                        /*c_mod=*/(short)0, C, /*reuse_a=*/false, /*reuse_b=*/false);
            }
            // C/D layout: VGPR v -> row M = v + 8*half, col N = lmod (doc-confirmed)
            #pragma unroll
            for (int v = 0; v < 8; ++v) {
                const int row = t * 16 + v + 8 * half;
                const int col = nt * 16 + lmod;
                if (row < PCONST && col < ACONST)
                    S[row * ACONST + col] = C[v];
            }
        }
    }

    __syncthreads();   // publish S across lanes (uniform barrier)

    // ---- Stage 4: banded 5-tap conv, fully-coalesced output stores ----
    // out[b, j] for j = 32*iter + lane : each store instr covers 128 contiguous bytes.
    if (active) {
        float* ob = out + (size_t)b * OUTSTRIDE;
        #pragma unroll
        for (int iter = 0; iter < OUTSTRIDE / 32; ++iter) {   // 17 iterations
            const int j = iter * 32 + lane;
            const int p = j >> 4;           // pocket row
            const int o = j & 15;           // output position
            const int aa = (int)xr[1 + LCONST * ACONST + p];
            const float* fw = kern + aa * FCONST;
            const float* Sp = S + p * ACONST + o;
            float acc = Sp[0] * fw[0];
            acc += Sp[1] * fw[1];
            acc += Sp[2] * fw[2];
            acc += Sp[3] * fw[3];
            acc += Sp[4] * fw[4];
            ob[j] = acc;
        }
    }
}

extern "C" void kernel_launch(void* const* d_in, const int* in_sizes, int n_in,
                              void* d_out, int out_size, void* d_ws, size_t ws_size,
                              hipStream_t stream) {
    const float* x     = (const float*)d_in[0];   // (B, 335) f32
    const float* kern  = (const float*)d_in[1];   // (20, 5)  f32
    const int*   table = (const int*)d_in[2];     // (16, 34, 15) i32
    float*       out   = (float*)d_out;           // (B, 544) f32

    const int nB = in_sizes[0] / XSTRIDE;
    dim3 block(256);
    dim3 grid((nB + WAVES_PER_BLOCK - 1) / WAVES_PER_BLOCK);
    hipLaunchKernelGGL(pep_pocket_conv_kernel, grid, block, 0, stream,
                       x, kern, table, out, nB);
}